// INRMLP_41377714930209
// MI455X (gfx1250) — compile-verified
//
#include <hip/hip_runtime.h>

// ---------------------------------------------------------------------------
// Batched SIREN INR-MLP (2 -> 20 -> 20 -> 20 -> 1, omega=20) on gfx1250.
// Compute-bound (~3.6 GFLOP vs ~27 MB): use fp32 matrix cores
// (v_wmma_f32_16x16x4_f32) to keep fp32 accuracy for sin(20*x) layers.
// One wave processes 16 points at a time:
//   D(16 outs x 16 pts) = A(W chunk 16x4) x B(acts 4x16) + C(bias)
// D stays in (feature x point) layout so layers chain via ds_bpermute
// re-striping (no LDS round trip for activations).
// ---------------------------------------------------------------------------

typedef float v2f __attribute__((ext_vector_type(2)));
typedef float v8f __attribute__((ext_vector_type(8)));

#define H       20
#define OMEGA   20.0f
#define TOTAL_P 921
#define W1_OFF  0
#define B1_OFF  40
#define W2_OFF  60
#define B2_OFF  460
#define W3_OFF  480
#define B3_OFF  880
#define W4_OFF  900
#define B4_OFF  920

#define NPTS            4096
#define NBATCH          512
#define NSPLIT          4
#define PTS_PER_BLOCK   (NPTS / NSPLIT)              // 1024
#define WAVES_PER_BLOCK 8
#define PTS_PER_WAVE    (PTS_PER_BLOCK / WAVES_PER_BLOCK) // 128
#define TILES_PER_WAVE  (PTS_PER_WAVE / 16)          // 8

__device__ __forceinline__ float bperm(float v, int srcLane) {
  return __builtin_bit_cast(
      float, __builtin_amdgcn_ds_bpermute(srcLane << 2,
                                          __builtin_bit_cast(int, v)));
}

__device__ __forceinline__ v8f wmma4(v2f a, v2f b, v8f c) {
  // D = A(16x4) * B(4x16) + C, fp32 throughout.
  return __builtin_amdgcn_wmma_f32_16x16x4_f32(
      /*neg_a=*/false, a, /*neg_b=*/false, b,
      /*c_mod=*/(short)0, c, /*reuse_a=*/false, /*reuse_b=*/false);
}

// Gather activation (feature k, point p) out of the D-layout accumulators
// h0 (feats 0..15) / h1 (feats 16..19). D layout: VGPR q holds M=q on lanes
// 0-15 and M=q+8 on lanes 16-31, N = lane%16.
__device__ __forceinline__ float gact(v8f h0, v8f h1, int k, int p) {
  const int r   = k & 15;
  const int src = p + ((r & 8) ? 16 : 0);
  const float v = (k < 16) ? h0[r & 7] : h1[r & 7];
  return bperm(v, src);
}

// Build the B fragment (4x16 f32) for K-chunk c from activations.
// B VGPR0 = row 4c (low lanes) / 4c+2 (high lanes); VGPR1 = 4c+1 / 4c+3.
__device__ __forceinline__ v2f makeB(v8f h0, v8f h1, int c, int p, bool low) {
  const float g0 = gact(h0, h1, 4 * c + 0, p);
  const float g1 = gact(h0, h1, 4 * c + 1, p);
  const float g2 = gact(h0, h1, 4 * c + 2, p);
  const float g3 = gact(h0, h1, 4 * c + 3, p);
  v2f b;
  b.x = low ? g0 : g2;
  b.y = low ? g1 : g3;
  return b;
}

__global__ __launch_bounds__(256)
void INRMLP_41377714930209_kernel(const float* __restrict__ coords,
                                  const float* __restrict__ flat_w,
                                  float* __restrict__ out) {
  __shared__ float sw[TOTAL_P];

  const int batch = blockIdx.x / NSPLIT;
  const int split = blockIdx.x % NSPLIT;
  const int tid   = threadIdx.x;

  // Stage this batch's 921 weights into LDS once.
  const float* wsrc = flat_w + (size_t)batch * TOTAL_P;
  for (int i = tid; i < TOTAL_P; i += 256) sw[i] = wsrc[i];
  __syncthreads();

  const int  lane = tid & 31;
  const int  wave = tid >> 5;
  const bool low  = lane < 16;
  const int  m    = lane & 15;      // row within A tile / point within B tile
  const int  khi  = low ? 0 : 2;    // K sub-offset per lane half

  // --- Weight A-fragments (A layout: VGPR v: K=v low lanes, K=v+2 high) ---
  v2f a1[2];            // layer1: 20x2, K padded to 4 (high lanes zero)
  v2f a2[2][5];         // layer2: 20x20, 5 K-chunks x 2 out-tiles
  v2f a3[2][5];         // layer3
  v2f a4[5];            // layer4: 1x20, out-tile 0 only (row 0)
  v8f cb1[2], cb2[2], cb3[2];

#pragma unroll
  for (int t = 0; t < 2; ++t) {
    const int  o  = t * 16 + m;
    const bool ov = o < H;
    a1[t].x = (ov && low) ? sw[W1_OFF + o * 2 + 0] : 0.f;
    a1[t].y = (ov && low) ? sw[W1_OFF + o * 2 + 1] : 0.f;
#pragma unroll
    for (int c = 0; c < 5; ++c) {
      const int kx = 4 * c + khi;
      a2[t][c].x = ov ? sw[W2_OFF + o * H + kx]     : 0.f;
      a2[t][c].y = ov ? sw[W2_OFF + o * H + kx + 1] : 0.f;
      a3[t][c].x = ov ? sw[W3_OFF + o * H + kx]     : 0.f;
      a3[t][c].y = ov ? sw[W3_OFF + o * H + kx + 1] : 0.f;
    }
    // Bias as C operand. C/D layout: VGPR q -> out (t*16 + q + 8*hi-half).
#pragma unroll
    for (int q = 0; q < 8; ++q) {
      const int idx = t * 16 + q + (low ? 0 : 8);
      const bool iv = idx < H;
      cb1[t][q] = iv ? sw[B1_OFF + idx] : 0.f;
      cb2[t][q] = iv ? sw[B2_OFF + idx] : 0.f;
      cb3[t][q] = iv ? sw[B3_OFF + idx] : 0.f;
    }
  }
#pragma unroll
  for (int c = 0; c < 5; ++c) {
    const int kx = 4 * c + khi;
    a4[c].x = (m == 0) ? sw[W4_OFF + kx]     : 0.f;
    a4[c].y = (m == 0) ? sw[W4_OFF + kx + 1] : 0.f;
  }
  v8f c4init = {};
  c4init[0] = low ? sw[B4_OFF] : 0.f;   // bias lands on row M=0 only

  // --- Point loop: each wave owns TILES_PER_WAVE tiles of 16 points ---
  const float2* cbase = (const float2*)(coords + (size_t)batch * NPTS * 2);
  float* obase        = out + (size_t)batch * NPTS;
  const int wave_pt0  = split * PTS_PER_BLOCK + wave * PTS_PER_WAVE;

  for (int tile = 0; tile < TILES_PER_WAVE; ++tile) {
    const int p0 = wave_pt0 + tile * 16;
    if (tile + 1 < TILES_PER_WAVE)
      __builtin_prefetch(&cbase[p0 + 16 + m], 0, 0);   // global_prefetch

    // Layer 1 input: B row0=x, row1=y, rows 2-3 zero-pad.
    const float2 xy = cbase[p0 + m];
    v2f bIn;
    bIn.x = low ? xy.x : 0.f;
    bIn.y = low ? xy.y : 0.f;

    v8f h[2];
#pragma unroll
    for (int t = 0; t < 2; ++t) h[t] = wmma4(a1[t], bIn, cb1[t]);
#pragma unroll
    for (int t = 0; t < 2; ++t)
#pragma unroll
      for (int q = 0; q < 8; ++q) h[t][q] = __sinf(OMEGA * h[t][q]);

    // Layer 2: accumulate 5 K-chunks into bias-initialized C.
    v8f h2[2] = {cb2[0], cb2[1]};
#pragma unroll
    for (int c = 0; c < 5; ++c) {
      const v2f bf = makeB(h[0], h[1], c, m, low);
      h2[0] = wmma4(a2[0][c], bf, h2[0]);
      h2[1] = wmma4(a2[1][c], bf, h2[1]);
    }
#pragma unroll
    for (int t = 0; t < 2; ++t)
#pragma unroll
      for (int q = 0; q < 8; ++q) h2[t][q] = __sinf(OMEGA * h2[t][q]);

    // Layer 3.
    v8f h3[2] = {cb3[0], cb3[1]};
#pragma unroll
    for (int c = 0; c < 5; ++c) {
      const v2f bf = makeB(h2[0], h2[1], c, m, low);
      h3[0] = wmma4(a3[0][c], bf, h3[0]);
      h3[1] = wmma4(a3[1][c], bf, h3[1]);
    }
#pragma unroll
    for (int t = 0; t < 2; ++t)
#pragma unroll
      for (int q = 0; q < 8; ++q) h3[t][q] = __sinf(OMEGA * h3[t][q]);

    // Layer 4: single output row; D row 0 = 16 points across lanes 0-15.
    v8f o4 = c4init;
#pragma unroll
    for (int c = 0; c < 5; ++c) {
      const v2f bf = makeB(h3[0], h3[1], c, m, low);
      o4 = wmma4(a4[c], bf, o4);
    }

    const float val = fminf(fmaxf(o4[0], 0.f), 1.f);
    if (low) obase[p0 + m] = val;   // coalesced 64B store
  }
}

extern "C" void kernel_launch(void* const* d_in, const int* in_sizes, int n_in,
                              void* d_out, int out_size, void* d_ws,
                              size_t ws_size, hipStream_t stream) {
  (void)in_sizes; (void)n_in; (void)out_size; (void)d_ws; (void)ws_size;
  const float* coords = (const float*)d_in[0];   // (512, 4096, 2) f32
  const float* flatw  = (const float*)d_in[1];   // (512, 921)     f32
  float* out          = (float*)d_out;           // (512, 4096, 1) f32

  dim3 grid(NBATCH * NSPLIT);
  dim3 block(256);
  INRMLP_41377714930209_kernel<<<grid, block, 0, stream>>>(coords, flatw, out);
}